// GCN_11854109737478
// MI455X (gfx1250) — compile-verified
//
#include <hip/hip_runtime.h>

constexpr int IN_DIM  = 128;
constexpr int OUT_DIM = 64;
constexpr int S1 = IN_DIM + 4;   // padded LDS column stride for W1^T (bank-conflict-free b64)
constexpr int S2 = OUT_DIM + 4;  // padded LDS column stride for W2^T

typedef float v2f __attribute__((ext_vector_type(2)));
typedef float v8f __attribute__((ext_vector_type(8)));

// ---------------------------------------------------------------- init ------
__global__ void __launch_bounds__(256) k_init(float* __restrict__ deg_s,
                                              float* __restrict__ deg_d,
                                              float* __restrict__ agg,
                                              float* __restrict__ hg, int n) {
    int tid = blockIdx.x * blockDim.x + threadIdx.x;
    int stride = gridDim.x * blockDim.x;
    for (int i = tid; i < n; i += stride) { deg_s[i] = 0.f; deg_d[i] = 0.f; }
    for (int i = tid; i < n * OUT_DIM; i += stride) agg[i] = 0.f;
    if (tid < 2 * OUT_DIM) hg[tid] = 0.f;
}

// ------------------------------------------------------------- degrees ------
__global__ void __launch_bounds__(256) k_degree(const int* __restrict__ src,
                                                const int* __restrict__ dst,
                                                float* __restrict__ outdeg,
                                                float* __restrict__ indeg, int E) {
    int e = blockIdx.x * blockDim.x + threadIdx.x;
    if (e < E) {
        atomicAdd(&outdeg[src[e]], 1.0f);
        atomicAdd(&indeg[dst[e]], 1.0f);
    }
}

__global__ void __launch_bounds__(256) k_norm(float* __restrict__ sn,
                                              float* __restrict__ dn, int n) {
    int i = blockIdx.x * blockDim.x + threadIdx.x;
    if (i < n) {
        sn[i] = rsqrtf(fmaxf(sn[i], 1.0f));
        dn[i] = rsqrtf(fmaxf(dn[i], 1.0f));
    }
}

// ------------------------------------------- GEMM1: t1 = (feat*srcn) @ W1 ---
// One wave per 16-row node tile; block = 8 waves. W1^T (64 cols x 128 k) in LDS.
__global__ void __launch_bounds__(256) k_gemm1(const float* __restrict__ feat,
                                               const float* __restrict__ W1,
                                               const float* __restrict__ srcn,
                                               float* __restrict__ t1, int n) {
    __shared__ float sWt[OUT_DIM * S1];                  // ~33 KB, transposed + padded
    for (int i = threadIdx.x; i < IN_DIM * OUT_DIM; i += 256) {
        const int k = i >> 6;      // row of W1 (k index)
        const int c = i & 63;      // column of W1
        sWt[c * S1 + k] = W1[i];
    }
    __syncthreads();

    const int wave = threadIdx.x >> 5;
    const int lane = threadIdx.x & 31;
    const int tile = blockIdx.x * 8 + wave;
    const int ntiles = (n + 15) >> 4;
    if (tile >= ntiles) return;

    const int halfsel = lane >> 4;       // 0: lanes 0-15 (K=0,1), 1: lanes 16-31 (K=2,3)
    const int l15 = lane & 15;
    int row = tile * 16 + l15;
    if (row >= n) row = n - 1;
    const int koff = halfsel * 2;
    const float norm = srcn[row];
    const float* arow = feat + (size_t)row * IN_DIM;
    const float* bc0 = sWt + ( 0 + l15) * S1;
    const float* bc1 = sWt + (16 + l15) * S1;
    const float* bc2 = sWt + (32 + l15) * S1;
    const float* bc3 = sWt + (48 + l15) * S1;

    v8f acc0 = {}, acc1 = {}, acc2 = {}, acc3 = {};
#pragma unroll 4
    for (int kb = 0; kb < IN_DIM / 4; ++kb) {
        const int k0 = kb * 4 + koff;            // always even -> 8B-aligned LDS b64
        v2f a = *(const v2f*)(arow + k0);
        a.x *= norm; a.y *= norm;
        const v2f b0 = *(const v2f*)(bc0 + k0);
        const v2f b1 = *(const v2f*)(bc1 + k0);
        const v2f b2 = *(const v2f*)(bc2 + k0);
        const v2f b3 = *(const v2f*)(bc3 + k0);
        acc0 = __builtin_amdgcn_wmma_f32_16x16x4_f32(false, a, false, b0, (short)0, acc0, false, false);
        acc1 = __builtin_amdgcn_wmma_f32_16x16x4_f32(false, a, false, b1, (short)0, acc1, false, false);
        acc2 = __builtin_amdgcn_wmma_f32_16x16x4_f32(false, a, false, b2, (short)0, acc2, false, false);
        acc3 = __builtin_amdgcn_wmma_f32_16x16x4_f32(false, a, false, b3, (short)0, acc3, false, false);
    }

    // C/D layout: VGPR r, lanes 0-15 -> (M=r, N=lane); lanes 16-31 -> (M=r+8, N=lane-16)
#pragma unroll
    for (int r = 0; r < 8; ++r) {
        const int node = tile * 16 + r + halfsel * 8;
        if (node < n) {
            float* o = t1 + (size_t)node * OUT_DIM + l15;
            o[0]  = acc0[r];
            o[16] = acc1[r];
            o[32] = acc2[r];
            o[48] = acc3[r];
        }
    }
}

// -------------------------------------------------------- edge scatter ------
// agg[dst[e]*64 + d..d+3] += msg[src[e]*64 + d..d+3], one thread per (edge, quad)
__global__ void __launch_bounds__(256) k_scatter(const int* __restrict__ src,
                                                 const int* __restrict__ dst,
                                                 const float* __restrict__ msg,
                                                 float* __restrict__ agg, int E) {
    int i = blockIdx.x * blockDim.x + threadIdx.x;
    if (i < E * (OUT_DIM / 4)) {
        const int e = i >> 4;
        const int q = (i & 15) << 2;
        const float4 m = *(const float4*)(msg + (size_t)src[e] * OUT_DIM + q);
        float* a = agg + (size_t)dst[e] * OUT_DIM + q;
        atomicAdd(a + 0, m.x);
        atomicAdd(a + 1, m.y);
        atomicAdd(a + 2, m.z);
        atomicAdd(a + 3, m.w);
    }
}

// ----------------------- finalize layer 1, emit hs = h1*srcn, pool hg1 ------
__global__ void __launch_bounds__(256) k_finalize1(float* __restrict__ agg,
                                                   const float* __restrict__ dstn,
                                                   const float* __restrict__ srcn,
                                                   const float* __restrict__ b1,
                                                   const float* __restrict__ prelu_a,
                                                   float* __restrict__ hs,
                                                   float* __restrict__ hg1, int n) {
    const float pa = *prelu_a;
    const int d = threadIdx.x & 63;
    const int sub = threadIdx.x >> 6;            // 4 nodes per block-iteration
    float part = 0.f;
    for (int base = blockIdx.x * 4; base < n; base += gridDim.x * 4) {
        const int node = base + sub;
        if (node < n) {
            const size_t idx = (size_t)node * OUT_DIM + d;
            const float v = agg[idx];
            agg[idx] = 0.f;                      // re-zero for layer-2 aggregation
            float h = v * dstn[node] + b1[d];
            h = (h >= 0.f) ? h : pa * h;
            part += h;
            hs[idx] = h * srcn[node];
        }
    }
    __shared__ float red[256];
    red[threadIdx.x] = part;
    __syncthreads();
    if (threadIdx.x < 128) red[threadIdx.x] += red[threadIdx.x + 128];
    __syncthreads();
    if (threadIdx.x < 64)
        atomicAdd(&hg1[threadIdx.x], red[threadIdx.x] + red[threadIdx.x + 64]);
}

// ------------------- GEMM2 + epilogue: h = prelu(agg2@W2 * dstn + b2) -------
__global__ void __launch_bounds__(256) k_gemm2(const float* __restrict__ agg,
                                               const float* __restrict__ W2,
                                               const float* __restrict__ dstn,
                                               const float* __restrict__ b2,
                                               const float* __restrict__ prelu_a,
                                               float* __restrict__ hout,
                                               float* __restrict__ hg2, int n) {
    __shared__ float sWt[OUT_DIM * S2];                  // ~17 KB, transposed + padded
    __shared__ float shg[OUT_DIM];
    for (int i = threadIdx.x; i < OUT_DIM * OUT_DIM; i += 256) {
        const int k = i >> 6;
        const int c = i & 63;
        sWt[c * S2 + k] = W2[i];
    }
    if (threadIdx.x < OUT_DIM) shg[threadIdx.x] = 0.f;
    __syncthreads();

    const float pa = *prelu_a;
    const int wave = threadIdx.x >> 5;
    const int lane = threadIdx.x & 31;
    const int halfsel = lane >> 4;
    const int l15 = lane & 15;
    const int tile = blockIdx.x * 8 + wave;
    const int ntiles = (n + 15) >> 4;
    const bool active = tile < ntiles;

    if (active) {
        int row = tile * 16 + l15;
        if (row >= n) row = n - 1;
        const int koff = halfsel * 2;
        const float* arow = agg + (size_t)row * OUT_DIM;
        const float* bc0 = sWt + ( 0 + l15) * S2;
        const float* bc1 = sWt + (16 + l15) * S2;
        const float* bc2 = sWt + (32 + l15) * S2;
        const float* bc3 = sWt + (48 + l15) * S2;

        v8f acc0 = {}, acc1 = {}, acc2 = {}, acc3 = {};
#pragma unroll 4
        for (int kb = 0; kb < OUT_DIM / 4; ++kb) {
            const int k0 = kb * 4 + koff;
            const v2f a  = *(const v2f*)(arow + k0);
            const v2f b0 = *(const v2f*)(bc0 + k0);
            const v2f b1 = *(const v2f*)(bc1 + k0);
            const v2f b2 = *(const v2f*)(bc2 + k0);
            const v2f b3 = *(const v2f*)(bc3 + k0);
            acc0 = __builtin_amdgcn_wmma_f32_16x16x4_f32(false, a, false, b0, (short)0, acc0, false, false);
            acc1 = __builtin_amdgcn_wmma_f32_16x16x4_f32(false, a, false, b1, (short)0, acc1, false, false);
            acc2 = __builtin_amdgcn_wmma_f32_16x16x4_f32(false, a, false, b2, (short)0, acc2, false, false);
            acc3 = __builtin_amdgcn_wmma_f32_16x16x4_f32(false, a, false, b3, (short)0, acc3, false, false);
        }

        const float bb0 = b2[ 0 + l15], bb1 = b2[16 + l15];
        const float bb2 = b2[32 + l15], bb3 = b2[48 + l15];
        float hsum0 = 0.f, hsum1 = 0.f, hsum2 = 0.f, hsum3 = 0.f;
#pragma unroll
        for (int r = 0; r < 8; ++r) {
            const int node = tile * 16 + r + halfsel * 8;
            if (node < n) {
                const float dn = dstn[node];
                float v0 = acc0[r] * dn + bb0; v0 = (v0 >= 0.f) ? v0 : pa * v0;
                float v1 = acc1[r] * dn + bb1; v1 = (v1 >= 0.f) ? v1 : pa * v1;
                float v2 = acc2[r] * dn + bb2; v2 = (v2 >= 0.f) ? v2 : pa * v2;
                float v3 = acc3[r] * dn + bb3; v3 = (v3 >= 0.f) ? v3 : pa * v3;
                float* o = hout + (size_t)node * OUT_DIM + l15;
                o[0] = v0; o[16] = v1; o[32] = v2; o[48] = v3;
                hsum0 += v0; hsum1 += v1; hsum2 += v2; hsum3 += v3;
            }
        }
        atomicAdd(&shg[ 0 + l15], hsum0);
        atomicAdd(&shg[16 + l15], hsum1);
        atomicAdd(&shg[32 + l15], hsum2);
        atomicAdd(&shg[48 + l15], hsum3);
    }
    __syncthreads();
    if (threadIdx.x < OUT_DIM) atomicAdd(&hg2[threadIdx.x], shg[threadIdx.x]);
}

// ---------------------------------------------------------------------------
extern "C" void kernel_launch(void* const* d_in, const int* in_sizes, int n_in,
                              void* d_out, int out_size, void* d_ws, size_t ws_size,
                              hipStream_t stream) {
    const float* feat    = (const float*)d_in[0];
    const int*   src     = (const int*)  d_in[1];
    const int*   dst     = (const int*)  d_in[2];
    const float* W1      = (const float*)d_in[3];
    const float* b1      = (const float*)d_in[4];
    const float* W2      = (const float*)d_in[5];
    const float* b2      = (const float*)d_in[6];
    const float* prelu_a = (const float*)d_in[7];

    const int n = in_sizes[0] / IN_DIM;     // 100000
    const int E = in_sizes[1];              // 1600000

    float* out = (float*)d_out;
    float* hg1 = out + (size_t)n * OUT_DIM; // pooled layer-1 sums
    float* hg2 = hg1 + OUT_DIM;             // pooled layer-2 sums

    float* wsf  = (float*)d_ws;
    float* srcn = wsf;                      // N
    float* dstn = wsf + n;                  // N
    float* agg  = wsf + 2 * (size_t)n;      // N*64 (agg1, re-zeroed in place, agg2)
    float* t1   = out;                      // reuse d_out h-region: t1 -> hs -> final h

    const int ntiles = (n + 15) / 16;
    const int gemm_blocks = (ntiles + 7) / 8;
    const long long tot = (long long)E * (OUT_DIM / 4);
    const int sblocks = (int)((tot + 255) / 256);

    k_init   <<<2048, 256, 0, stream>>>(srcn, dstn, agg, hg1, n);
    k_degree <<<(E + 255) / 256, 256, 0, stream>>>(src, dst, srcn, dstn, E);
    k_norm   <<<(n + 255) / 256, 256, 0, stream>>>(srcn, dstn, n);
    k_gemm1  <<<gemm_blocks, 256, 0, stream>>>(feat, W1, srcn, t1, n);
    k_scatter<<<sblocks, 256, 0, stream>>>(src, dst, t1, agg, E);
    k_finalize1<<<1024, 256, 0, stream>>>(agg, dstn, srcn, b1, prelu_a, t1, hg1, n);
    k_scatter<<<sblocks, 256, 0, stream>>>(src, dst, t1, agg, E);
    k_gemm2  <<<gemm_blocks, 256, 0, stream>>>(agg, W2, dstn, b2, prelu_a, out, hg2, n);
}